// RandomSparseAttention_36206574306109
// MI455X (gfx1250) — compile-verified
//
#include <hip/hip_runtime.h>
#include <hip/hip_bf16.h>

// Problem constants (match reference)
#define B_ 4
#define S_ 4096
#define D_ 512
#define NR_ 32
#define SCALE_ 0.044194173824159216f  // 1/sqrt(512)

typedef __attribute__((ext_vector_type(2))) float v2f;
typedef __attribute__((ext_vector_type(8))) float v8f;

// ---------------------------------------------------------------------------
// Pass 1: compact mask rows into column-index lists (<=32 per row) and emit
// the mask-as-float output region. One wave (32 lanes) per row.
// ---------------------------------------------------------------------------
__global__ __launch_bounds__(128) void mask_prep_kernel(
    const unsigned char* __restrict__ mask, float* __restrict__ maskf,
    int* __restrict__ idx, int* __restrict__ cnt) {
  const int row  = blockIdx.x * (blockDim.x >> 5) + (threadIdx.x >> 5);
  const int lane = threadIdx.x & 31;
  if (row >= S_) return;

  const unsigned char* mrow = mask + (size_t)row * S_;
  float* frow = maskf + (size_t)row * S_;
  int base = 0;
  const unsigned lt_mask = (1u << lane) - 1u;

  for (int c0 = 0; c0 < S_; c0 += 32) {
    const bool m = mrow[c0 + lane] != 0;
    frow[c0 + lane] = m ? 1.0f : 0.0f;
    const unsigned bal = __builtin_amdgcn_ballot_w32(m);
    if (m) {
      const int pos = base + __popc(bal & lt_mask);
      if (pos < NR_) idx[row * NR_ + pos] = c0 + lane;
    }
    base += __popc(bal);
  }
  if (lane == 0) cnt[row] = (base < NR_) ? base : NR_;
}

// ---------------------------------------------------------------------------
// Pass 2: zero the dense attn output region (grid-stride float4).
// ---------------------------------------------------------------------------
__global__ __launch_bounds__(256) void zero_attn_kernel(float4* __restrict__ p,
                                                        long long n4) {
  const long long stride = (long long)gridDim.x * blockDim.x;
  for (long long i = (long long)blockIdx.x * blockDim.x + threadIdx.x; i < n4;
       i += stride)
    p[i] = make_float4(0.f, 0.f, 0.f, 0.f);
}

// ---------------------------------------------------------------------------
// Pass 3: sparse attention, one wave per (b, row).
//   Scores via V_WMMA_F32_16X16X4_F32:
//     A (16x4 f32): q[d0..d0+3] broadcast over M. Layout: lanes 0-15 hold
//       {K=0,K=1}, lanes 16-31 hold {K=2,K=3}  -> v2f = q[d0+2*hi .. +1].
//     B (4x16 f32): lane n supplies gathered column col[n]'s chunk:
//       lanes 0-15 {K=0,K=1}, lanes 16-31 {K=2,K=3} -> K[col][d0+2*hi .. +1].
//     C replicates scores over M: C[0] on lane L = score[(L&15) + 16*tile].
// ---------------------------------------------------------------------------
#define WPB 8  // waves per block
__global__ __launch_bounds__(32 * WPB) void attn_kernel(
    const float* __restrict__ q, const float* __restrict__ k,
    const float* __restrict__ v, const int* __restrict__ idx,
    const int* __restrict__ cnt, float* __restrict__ outp,
    float* __restrict__ attnp) {
  __shared__ __align__(16) float qs_all[WPB * D_];

  const int wave = threadIdx.x >> 5;
  const int lane = threadIdx.x & 31;
  const int task = blockIdx.x * WPB + wave;  // == b*S_ + row
  const int b    = task >> 12;               // / S_
  const int row  = task & (S_ - 1);

  // Stage q row into LDS (128B-coalesced float4)
  float* qs = qs_all + wave * D_;
  {
    const float4* q4  = (const float4*)(q + (size_t)task * D_);
    float4*       qs4 = (float4*)qs;
#pragma unroll
    for (int t = 0; t < 4; ++t) qs4[lane + 32 * t] = q4[lane + 32 * t];
  }
  __syncthreads();

  const int n   = cnt[row];  // uniform per wave (scalar)
  const int col = (lane < n) ? idx[row * NR_ + lane] : row;  // padded: diag

  // B-matrix column indices this lane must supply for the two 16-col tiles
  const int colA = __shfl(col, lane & 15, 32);
  const int colB = __shfl(col, (lane & 15) + 16, 32);
  const int hi   = lane >> 4;  // 0: K=0,1   1: K=2,3

  const float*  qsv = qs + 2 * hi;
  const float2* pA  = (const float2*)(k + ((size_t)b * S_ + colA) * D_ + 2 * hi);
  const float2* pB  = (const float2*)(k + ((size_t)b * S_ + colB) * D_ + 2 * hi);

  v8f c0 = {}, c1 = {};
#pragma unroll 4
  for (int d0 = 0; d0 < D_; d0 += 4) {
    const float2 av  = *(const float2*)(qsv + d0);
    const float2 b0v = pA[d0 >> 1];
    const float2 b1v = pB[d0 >> 1];
    v2f a;  a.x = av.x;  a.y = av.y;
    v2f b0; b0.x = b0v.x; b0.y = b0v.y;
    v2f b1; b1.x = b1v.x; b1.y = b1v.y;
    c0 = __builtin_amdgcn_wmma_f32_16x16x4_f32(false, a, false, b0, (short)0,
                                               c0, false, false);
    c1 = __builtin_amdgcn_wmma_f32_16x16x4_f32(false, a, false, b1, (short)0,
                                               c1, false, false);
  }

  // Lane j's score (scores replicated over M, so element 0 always works)
  float p = (hi ? c1[0] : c0[0]) * SCALE_;
  if (lane >= n) p = -__builtin_inff();

  // Wave-wide softmax
  float m = p;
#pragma unroll
  for (int o = 16; o > 0; o >>= 1) m = fmaxf(m, __shfl_xor(m, o, 32));
  float e = (lane < n) ? __expf(p - m) : 0.0f;
  float sum = e;
#pragma unroll
  for (int o = 16; o > 0; o >>= 1) sum += __shfl_xor(sum, o, 32);
  const float pr = e / sum;

  // Scattered attn writes (<=32 per row; rest of the row stays zero)
  if (lane < n) attnp[(size_t)task * S_ + col] = pr;

  // P·V: lane owns 16 contiguous d-values (4x float4), coalesced across wave
  float4 acc[4];
#pragma unroll
  for (int t = 0; t < 4; ++t) acc[t] = make_float4(0.f, 0.f, 0.f, 0.f);

  for (int j = 0; j < n; ++j) {
    const float pj = __shfl(pr, j, 32);
    const int   cj = __shfl(col, j, 32);
    const float4* vp =
        (const float4*)(v + ((size_t)b * S_ + cj) * D_) + lane * 4;
    if (j + 1 < n) {
      const int cn = __shfl(col, j + 1, 32);
      __builtin_prefetch(v + ((size_t)b * S_ + cn) * D_ + lane * 16, 0, 1);
    }
#pragma unroll
    for (int t = 0; t < 4; ++t) {
      const float4 vv = vp[t];
      acc[t].x += pj * vv.x;
      acc[t].y += pj * vv.y;
      acc[t].z += pj * vv.z;
      acc[t].w += pj * vv.w;
    }
  }

  float4* op = (float4*)(outp + (size_t)task * D_) + lane * 4;
#pragma unroll
  for (int t = 0; t < 4; ++t) op[t] = acc[t];
}

// ---------------------------------------------------------------------------
extern "C" void kernel_launch(void* const* d_in, const int* in_sizes, int n_in,
                              void* d_out, int out_size, void* d_ws,
                              size_t ws_size, hipStream_t stream) {
  const float* q = (const float*)d_in[0];
  const float* k = (const float*)d_in[1];
  const float* v = (const float*)d_in[2];
  const unsigned char* mask = (const unsigned char*)d_in[3];  // jax bool = u8

  float* out   = (float*)d_out;
  float* attnp = out + (size_t)B_ * S_ * D_;
  float* maskf = attnp + (size_t)B_ * S_ * S_;

  int* idx  = (int*)d_ws;        // S_*NR_ ints
  int* cntb = idx + S_ * NR_;    // S_ ints

  // Pass 1: compact mask + emit mask-as-float region (one wave per row)
  mask_prep_kernel<<<S_ / 4, 128, 0, stream>>>(mask, maskf, idx, cntb);

  // Pass 2: zero dense attn region
  const long long n4 = (long long)B_ * S_ * S_ / 4;
  zero_attn_kernel<<<4096, 256, 0, stream>>>((float4*)attnp, n4);

  // Pass 3: sparse attention (one wave per (b,row); 8 waves per block)
  attn_kernel<<<(B_ * S_) / WPB, 32 * WPB, 0, stream>>>(q, k, v, idx, cntb,
                                                        out, attnp);
}